// ExampleNet_18760417149163
// MI455X (gfx1250) — compile-verified
//
#include <hip/hip_runtime.h>
#include <cstdint>
#include <cstddef>

#define GD   160
#define GN   (GD * GD * GD)
#define CIN  32
#define COUT 64

typedef __attribute__((ext_vector_type(2))) float v2f;
typedef __attribute__((ext_vector_type(4))) float v4f;
typedef __attribute__((ext_vector_type(8))) float v8f;
typedef __attribute__((ext_vector_type(4))) int   v4i;

// CDNA5 async global->LDS path (gfx1250): guarded feature detection.
#if defined(__gfx1250__) &&                                          \
    __has_builtin(__builtin_amdgcn_global_load_async_to_lds_b128) && \
    __has_builtin(__builtin_amdgcn_s_wait_asynccnt)
#define USE_ASYNC_LDS 1
#else
#define USE_ASYNC_LDS 0
#endif

// Builtin signature (from hipcc diagnostic): typed int4 pointers,
// AS(1) global source ("__device__"), AS(3) LDS destination ("__shared__").
#define AS1_B128(p) ((__attribute__((address_space(1))) v4i*)(p))
#define AS3_B128(p) ((__attribute__((address_space(3))) v4i*)(p))

// ---------------- grid build ----------------
__global__ void grid_init_kernel(int* __restrict__ grid) {
  int i = blockIdx.x * blockDim.x + threadIdx.x;
  if (i < GN) grid[i] = -1;
}

__global__ void grid_scatter_kernel(const int* __restrict__ coors,
                                    int* __restrict__ grid, int n) {
  int i = blockIdx.x * blockDim.x + threadIdx.x;
  if (i < n) {
    int z = coors[i * 4 + 1];
    int y = coors[i * 4 + 2];
    int x = coors[i * 4 + 3];
    grid[(z * GD + y) * GD + x] = i;
  }
}

// ---------------- sparse conv via WMMA ----------------
// Block: 128 threads (4 wave32s). M-tile = 16 voxels.
// Wave w computes output columns [16w, 16w+16) via 27*8 V_WMMA_F32_16X16X4_F32.
// Staging of A (gathered features) and W_k uses GLOBAL_LOAD_ASYNC_TO_LDS_B128
// when available (ASYNCcnt-tracked), else b128 load + ds_store.
__launch_bounds__(128)
__global__ void spconv_wmma_kernel(const float* __restrict__ feats,
                                   const int* __restrict__ coors,
                                   const float* __restrict__ weight,
                                   const float* __restrict__ bias,
                                   const int* __restrict__ grid,
                                   float* __restrict__ out,
                                   int nvox) {
  __shared__ __align__(16) float Atile[16 * CIN];    // 2 KB gathered features
  __shared__ __align__(16) float Wtile[CIN * COUT];  // 8 KB weight tap
  __shared__ int scoor[16 * 3];

  const int t    = threadIdx.x;
  const int lane = t & 31;
  const int hi   = lane >> 4;   // half-wave select
  const int wv   = t >> 5;      // wave id 0..3 -> N-slice
  const int baseM = blockIdx.x << 4;

  // stage this tile's voxel coords
  if (t < 16) {
    int m = baseM + t;
    if (m >= nvox) m = nvox - 1;
    const int* c = coors + (size_t)m * 4;
    scoor[t * 3 + 0] = c[1];
    scoor[t * 3 + 1] = c[2];
    scoor[t * 3 + 2] = c[3];
  }
  __syncthreads();

  // gather assignment: thread t loads row (t>>3), channels 4*(t&7)..+4
  const int grow = t >> 3;
  const int gch  = t & 7;
  const int zz = scoor[grow * 3 + 0];
  const int yy = scoor[grow * 3 + 1];
  const int xx = scoor[grow * 3 + 2];

  const int ncol = (wv << 4) + (lane & 15);  // output column in [0,64)

  v8f acc;
  {
    float bv = bias[ncol];
#pragma unroll
    for (int i = 0; i < 8; ++i) acc[i] = bv;
  }

  for (int k = 0; k < 27; ++k) {
    // prefetch next tap's weights toward L2 (global_prefetch_b8)
    {
      int kn = (k + 1 < 27) ? (k + 1) : 0;
      __builtin_prefetch(weight + (size_t)kn * (CIN * COUT) + t * 16, 0, 1);
    }
    __syncthreads();  // WAR: previous iteration's LDS reads complete

    // neighbor lookup for this thread's gather row
    const int dz = k / 9 - 1, dy = (k / 3) % 3 - 1, dx = k % 3 - 1;
    const int nz = zz + dz, ny = yy + dy, nx = xx + dx;
    int nidx = -1;
    if ((unsigned)nz < (unsigned)GD && (unsigned)ny < (unsigned)GD &&
        (unsigned)nx < (unsigned)GD)
      nidx = grid[(nz * GD + ny) * GD + nx];

    float* adst = &Atile[grow * CIN + gch * 4];
    const float* wsrc = weight + (size_t)k * (CIN * COUT) + t * 16;
    float* wdst = &Wtile[t * 16];

#if USE_ASYNC_LDS
    // A tile: valid lanes stream directly global->LDS; invalid lanes zero-fill.
    if (nidx >= 0) {
      __builtin_amdgcn_global_load_async_to_lds_b128(
          AS1_B128(feats + (size_t)nidx * CIN + gch * 4), AS3_B128(adst), 0, 0);
    } else {
      *(v4f*)adst = (v4f){0.f, 0.f, 0.f, 0.f};
    }
    // W_k tile: 2048 dwords, 16 per thread (4 x async b128)
#pragma unroll
    for (int j = 0; j < 4; ++j)
      __builtin_amdgcn_global_load_async_to_lds_b128(
          AS1_B128(wsrc + j * 4), AS3_B128(wdst + j * 4), 0, 0);
    __builtin_amdgcn_s_wait_asynccnt(0);
#else
    v4f av = (v4f){0.f, 0.f, 0.f, 0.f};
    if (nidx >= 0)
      av = *(const v4f*)(feats + (size_t)nidx * CIN + gch * 4);
    *(v4f*)adst = av;
#pragma unroll
    for (int j = 0; j < 4; ++j)
      *(v4f*)(wdst + j * 4) = *(const v4f*)(wsrc + j * 4);
#endif

    __syncthreads();

    // 8 K-steps of 16x16x4 fp32 WMMA over this tap's K=32
    // A 16x4 layout: lane M = lane%16; lanes<16 hold K={0,1}, lanes>=16 K={2,3}
    const float* abase = &Atile[(lane & 15) * CIN + (hi << 1)];
#pragma unroll
    for (int s = 0; s < 8; ++s) {
      v2f a = *(const v2f*)(abase + 4 * s);
      const int kb = 4 * s + (hi << 1);
      v2f b;  // B 4x16: VGPR j holds K=j (lanes<16) / K=j+2 (lanes>=16)
      b.x = Wtile[kb * COUT + ncol];
      b.y = Wtile[(kb + 1) * COUT + ncol];
      acc = __builtin_amdgcn_wmma_f32_16x16x4_f32(
          false, a, false, b, (short)0, acc, false, false);
    }
  }

  // C/D layout: VGPR i -> M = i (lanes<16) / M = 8+i (lanes>=16)
  const int rbase = baseM + (hi << 3);
#pragma unroll
  for (int i = 0; i < 8; ++i) {
    int m = rbase + i;
    if (m < nvox)
      out[(size_t)m * COUT + ncol] = acc[i];
  }
}

// ---------------- launch ----------------
extern "C" void kernel_launch(void* const* d_in, const int* in_sizes, int n_in,
                              void* d_out, int out_size, void* d_ws, size_t ws_size,
                              hipStream_t stream) {
  const float* feats  = (const float*)d_in[0];
  const int*   coors  = (const int*)d_in[1];
  const float* weight = (const float*)d_in[2];
  const float* bias   = (const float*)d_in[3];
  float* out = (float*)d_out;

  const int nvox = in_sizes[0] / CIN;
  int* grid = (int*)d_ws;  // 160^3 * 4B = 16.4 MB scratch

  grid_init_kernel<<<(GN + 255) / 256, 256, 0, stream>>>(grid);
  grid_scatter_kernel<<<(nvox + 255) / 256, 256, 0, stream>>>(coors, grid, nvox);

  const int mtiles = (nvox + 15) / 16;
  spconv_wmma_kernel<<<mtiles, 128, 0, stream>>>(feats, coors, weight, bias,
                                                 grid, out, nvox);
}